// MultiHeadGATLayer_84344567759053
// MI455X (gfx1250) — compile-verified
//
#include <hip/hip_runtime.h>
#include <hip/hip_bf16.h>

typedef __attribute__((ext_vector_type(2))) float v2f;
typedef __attribute__((ext_vector_type(4))) float v4f;
typedef __attribute__((ext_vector_type(8))) float v8f;

#define N_NODES  50000
#define N_EDGES  800000
#define IN_F     256
#define OUT_F    64
#define N_HEADS  8
#define COLS     (N_HEADS * OUT_F)   // 512
#define ROW_TILES (N_NODES / 16)     // 3125 (exact)
#define LRELU_SLOPE 0.2f

// ---------------------------------------------------------------------------
// Zero-fill (d_ws / d_out are poisoned with 0xAA; we must init accumulators)
// ---------------------------------------------------------------------------
__global__ void zero_f32(float* __restrict__ p, int n) {
    int i = blockIdx.x * blockDim.x + threadIdx.x;
    if (i < n) p[i] = 0.0f;
}

// ---------------------------------------------------------------------------
// Phase 1: h[n, head*64+f] = sum_k x[n,k] * W[head,k,f]
//
// Block = 8 waves, one head per block (blockIdx.x), 8 row tiles per block
// (one per wave). W[head] (256x64 f32 = 64KB) is staged in LDS once per
// block; each wave computes a 16x64 strip with 4 f32 accumulators, so one
// A-operand load (b64) feeds 4 V_WMMA_F32_16X16X4_F32 ops and the B stream
// comes from LDS. Global load traffic: 2 B/lane per WMMA (vs 16 naive).
// ---------------------------------------------------------------------------
__global__ __launch_bounds__(256) void gat_gemm_wmma(const float* __restrict__ x,
                                                     const float* __restrict__ W,
                                                     float* __restrict__ h) {
    __shared__ float Wlds[IN_F * OUT_F];            // 64 KB

    const int wave = threadIdx.x >> 5;
    const int lane = threadIdx.x & 31;
    const int head = blockIdx.x;                    // 0..7

    // --- stage W[head] into LDS (coalesced b128 copies: 4096 float4 / 256 thr)
    {
        const v4f* __restrict__ Wg = (const v4f*)(W + (size_t)head * (IN_F * OUT_F));
        v4f* __restrict__ Wl = (v4f*)Wlds;
#pragma unroll
        for (int i = 0; i < 16; ++i) {
            int t = threadIdx.x + i * 256;
            Wl[t] = Wg[t];
        }
    }
    __syncthreads();

    const int rowTile = blockIdx.y * 8 + wave;      // 0..3124
    if (rowTile >= ROW_TILES) return;               // wave-uniform (EXEC all-1s below)

    const int row0 = rowTile * 16;
    const int fo   = lane & 15;                     // column-in-tile
    // A-operand: lanes 0-15 -> K = k0+0/+1 ; lanes 16-31 -> K = k0+2/+3
    const int koff = (lane >> 4) << 1;              // 0 or 2
    const int arow = row0 + (lane & 15);
    const float* __restrict__ ap = x + (size_t)arow * IN_F + koff;

    v8f acc0 = {0.f,0.f,0.f,0.f,0.f,0.f,0.f,0.f};
    v8f acc1 = acc0, acc2 = acc0, acc3 = acc0;

#pragma unroll 2
    for (int k0 = 0; k0 < IN_F; k0 += 4) {
        v2f a = *(const v2f*)(ap + k0);             // 8 B/lane global
        const float* __restrict__ b0 = Wlds + (k0 + koff) * OUT_F + fo;
        const float* __restrict__ b1 = b0 + OUT_F;  // next K row
        v2f b;
        b.x = b0[0];  b.y = b1[0];
        acc0 = __builtin_amdgcn_wmma_f32_16x16x4_f32(false, a, false, b,
                                                     (short)0, acc0, false, false);
        b.x = b0[16]; b.y = b1[16];
        acc1 = __builtin_amdgcn_wmma_f32_16x16x4_f32(false, a, false, b,
                                                     (short)0, acc1, false, false);
        b.x = b0[32]; b.y = b1[32];
        acc2 = __builtin_amdgcn_wmma_f32_16x16x4_f32(false, a, false, b,
                                                     (short)0, acc2, false, false);
        b.x = b0[48]; b.y = b1[48];
        acc3 = __builtin_amdgcn_wmma_f32_16x16x4_f32(false, a, false, b,
                                                     (short)0, acc3, false, false);
    }

    // C/D layout: VGPR v, lanes 0-15 -> M=v ; lanes 16-31 -> M=v+8
    const int rbase = row0 + ((lane >> 4) << 3);
    float* __restrict__ hp = h + head * OUT_F + fo;
#pragma unroll
    for (int v = 0; v < 8; ++v) {
        float* __restrict__ hr = hp + (size_t)(rbase + v) * COLS;
        hr[0]  = acc0[v];
        hr[16] = acc1[v];
        hr[32] = acc2[v];
        hr[48] = acc3[v];
    }
}

// ---------------------------------------------------------------------------
// Phase 1b: per-(node,head) attention coefficients
//   a_self[n,h] = h[n,h,:] . a[h,0,:]   a_neigh[n,h] = h[n,h,:] . a[h,1,:]
// ---------------------------------------------------------------------------
__global__ void gat_attn_coef(const float* __restrict__ h,
                              const float* __restrict__ a,
                              float* __restrict__ a_self,
                              float* __restrict__ a_neigh) {
    int idx = blockIdx.x * blockDim.x + threadIdx.x;   // N_NODES * N_HEADS
    if (idx >= N_NODES * N_HEADS) return;
    int n = idx >> 3;
    int hd = idx & 7;
    const v4f* __restrict__ hp = (const v4f*)(h + (size_t)n * COLS + hd * OUT_F);
    const v4f* __restrict__ as = (const v4f*)(a + (size_t)hd * 2 * OUT_F);
    const v4f* __restrict__ an = as + (OUT_F / 4);
    float s = 0.f, t = 0.f;
#pragma unroll
    for (int f = 0; f < OUT_F / 4; ++f) {
        v4f v = hp[f], u = as[f], w = an[f];
        s += v.x * u.x + v.y * u.y + v.z * u.z + v.w * u.w;
        t += v.x * w.x + v.y * w.y + v.z * w.z + v.w * w.w;
    }
    a_self[idx]  = s;
    a_neigh[idx] = t;
}

__device__ __forceinline__ float edge_exp(float as, float an, float m) {
    float lg = as + an;
    lg = (lg > 0.f ? lg : LRELU_SLOPE * lg) * m;
    return __expf(lg);
}

// ---------------------------------------------------------------------------
// Phase 2a: softmax denominators  denom[dst,h] += exp(e)
// One thread per (edge, head); a_self/a_neigh/denom are L2-resident (1.6 MB).
// ---------------------------------------------------------------------------
__global__ void gat_edge_denom(const int* __restrict__ src,
                               const int* __restrict__ dst,
                               const float* __restrict__ M,
                               const float* __restrict__ a_self,
                               const float* __restrict__ a_neigh,
                               float* __restrict__ denom) {
    int idx = blockIdx.x * blockDim.x + threadIdx.x;   // N_EDGES * N_HEADS
    if (idx >= N_EDGES * N_HEADS) return;
    int e  = idx >> 3;
    int hd = idx & 7;
    int s = src[e], d = dst[e];
    float ex = edge_exp(a_self[s * N_HEADS + hd], a_neigh[d * N_HEADS + hd], M[e]);
    atomicAdd(&denom[d * N_HEADS + hd], ex);
}

// ---------------------------------------------------------------------------
// Phase 2b: out[dst,h,:] += (exp(e)/denom[dst,h]) * h[src,h,:]
// One wave per edge; lanes span the feature dim -> 128B coalesced gathers of
// h[src] and coalesced f32 atomics into out[dst]. exp(e) is recomputed
// (2 VALU ops) instead of spilling a 25.6MB exp_e buffer.
// ---------------------------------------------------------------------------
__global__ __launch_bounds__(256) void gat_edge_aggregate(const int* __restrict__ src,
                                                          const int* __restrict__ dst,
                                                          const float* __restrict__ M,
                                                          const float* __restrict__ a_self,
                                                          const float* __restrict__ a_neigh,
                                                          const float* __restrict__ denom,
                                                          const float* __restrict__ h,
                                                          float* __restrict__ out) {
    const int wave = threadIdx.x >> 5;
    const int lane = threadIdx.x & 31;
    const int e = blockIdx.x * 8 + wave;
    if (e >= N_EDGES) return;

    const int s = src[e];
    const int d = dst[e];
    const float m = M[e];
    const float* __restrict__ hs = h   + (size_t)s * COLS;
    float* __restrict__       od = out + (size_t)d * COLS;

#pragma unroll
    for (int hd = 0; hd < N_HEADS; ++hd) {
        float ex  = edge_exp(a_self[s * N_HEADS + hd], a_neigh[d * N_HEADS + hd], m);
        float att = ex / (denom[d * N_HEADS + hd] + 1e-16f);
        int f0 = hd * OUT_F + lane;
        atomicAdd(&od[f0],      att * hs[f0]);
        atomicAdd(&od[f0 + 32], att * hs[f0 + 32]);
    }
}

// ---------------------------------------------------------------------------
// Phase 3: in-place ELU on the accumulated output
// ---------------------------------------------------------------------------
__global__ void gat_elu(float* __restrict__ out, int n) {
    int i = blockIdx.x * blockDim.x + threadIdx.x;
    if (i < n) {
        float v = out[i];
        out[i] = v > 0.f ? v : (__expf(v) - 1.0f);
    }
}

// ---------------------------------------------------------------------------
extern "C" void kernel_launch(void* const* d_in, const int* in_sizes, int n_in,
                              void* d_out, int out_size, void* d_ws, size_t ws_size,
                              hipStream_t stream) {
    const float* x   = (const float*)d_in[0];   // [N, 256]
    const int*   src = (const int*)  d_in[1];   // [E]
    const int*   dst = (const int*)  d_in[2];   // [E]
    const float* M   = (const float*)d_in[3];   // [E]
    const float* W   = (const float*)d_in[4];   // [8, 256, 64]
    const float* a   = (const float*)d_in[5];   // [8, 2, 64]
    float* out = (float*)d_out;                  // [N, 512]

    // Workspace layout: h | a_self | a_neigh | denom
    float* h       = (float*)d_ws;                          // N*512
    float* a_self  = h      + (size_t)N_NODES * COLS;       // N*8
    float* a_neigh = a_self + (size_t)N_NODES * N_HEADS;    // N*8
    float* denom   = a_neigh+ (size_t)N_NODES * N_HEADS;    // N*8

    // Init accumulators (out and denom)
    {
        int n = N_NODES * COLS;
        zero_f32<<<(n + 255) / 256, 256, 0, stream>>>(out, n);
        int m = N_NODES * N_HEADS;
        zero_f32<<<(m + 255) / 256, 256, 0, stream>>>(denom, m);
    }

    // Phase 1: dense fp32 WMMA GEMM -> h  (one head per block.x)
    {
        dim3 grid(N_HEADS, (ROW_TILES + 7) / 8);
        gat_gemm_wmma<<<grid, 256, 0, stream>>>(x, W, h);
    }

    // Phase 1b: attention coefficients
    {
        int n = N_NODES * N_HEADS;
        gat_attn_coef<<<(n + 255) / 256, 256, 0, stream>>>(h, a, a_self, a_neigh);
    }

    // Phase 2a: softmax denominators
    {
        int n = N_EDGES * N_HEADS;
        gat_edge_denom<<<(n + 255) / 256, 256, 0, stream>>>(src, dst, M,
                                                            a_self, a_neigh, denom);
    }

    // Phase 2b: attention-weighted aggregation (one wave per edge)
    {
        int blocks = (N_EDGES + 7) / 8;
        gat_edge_aggregate<<<blocks, 256, 0, stream>>>(src, dst, M, a_self, a_neigh,
                                                       denom, h, out);
    }

    // Phase 3: ELU
    {
        int n = N_NODES * COLS;
        gat_elu<<<(n + 255) / 256, 256, 0, stream>>>(out, n);
    }
}